// GATWithEdgeEmbedding_1Layer_62251255988695
// MI455X (gfx1250) — compile-verified
//
#include <hip/hip_runtime.h>
#include <hip/hip_bf16.h>

#define NODE_DIM 128
#define EDGE_DIM 64
#define NEG_SLOPE 0.2f

typedef __attribute__((ext_vector_type(2))) float v2f;
typedef __attribute__((ext_vector_type(8))) float v8f;

__device__ __forceinline__ v8f wmma_f32_4(v2f a, v2f b, v8f c) {
  // D = A(16x4 f32) * B(4x16 f32) + C(16x16 f32)
  return __builtin_amdgcn_wmma_f32_16x16x4_f32(
      /*neg_a=*/false, a, /*neg_b=*/false, b,
      /*c_mod=*/(short)0, c, /*reuse_a=*/false, /*reuse_b=*/false);
}

__device__ __forceinline__ float leaky(float v) {
  return v > 0.f ? v : NEG_SLOPE * v;
}

// float atomic max via ordered-int trick (works for mixed signs; init with -inf)
__device__ __forceinline__ void atomicMaxF(float* addr, float val) {
  if (val >= 0.f) atomicMax((int*)addr, __float_as_int(val));
  else            atomicMin((unsigned int*)addr, __float_as_uint(val));
}

// ---------------------------------------------------------------- init kernels
__global__ void k_zero(float* p, long n) {
  long i = (long)blockIdx.x * blockDim.x + threadIdx.x;
  if (i < n) p[i] = 0.f;
}

__global__ void k_init_node(float* cnt, float* segmax, float* den, int N) {
  int i = blockIdx.x * blockDim.x + threadIdx.x;
  if (i < N) { cnt[i] = 0.f; segmax[i] = -INFINITY; den[i] = 0.f; }
}

// ------------------------------------------------- edge GEMM + scatter (WMMA)
// ee = edge_attr @ We + be, atomically accumulated into sums[dst]
__global__ void k_edge_embed(const float* __restrict__ edge_attr,
                             const long long* __restrict__ eidx,
                             const float* __restrict__ We,
                             const float* __restrict__ be,
                             float* __restrict__ sums, int E) {
  __shared__ float sWe[EDGE_DIM * NODE_DIM];  // 32 KB
  for (int i = threadIdx.x; i < EDGE_DIM * NODE_DIM; i += blockDim.x)
    sWe[i] = We[i];
  __syncthreads();

  const int lane = threadIdx.x & 31;
  const int wave = threadIdx.x >> 5;
  const int wpb  = blockDim.x >> 5;
  const int half = lane >> 4;   // 0: lanes 0-15 (K0,K1) ; 1: lanes 16-31 (K2,K3)
  const int l16  = lane & 15;
  const int nTiles = (E + 15) >> 4;

  // hoist bias: lane only ever needs be[nt*16 + l16]
  float bev[8];
  #pragma unroll
  for (int nt = 0; nt < 8; ++nt) bev[nt] = be[nt * 16 + l16];

  for (int t = blockIdx.x * wpb + wave; t < nTiles; t += gridDim.x * wpb) {
    const int  e0   = t << 4;
    const bool full = (e0 + 16) <= E;  // wave-uniform: tile fully in range

    // A fragments: 16 K-steps of 16x4; lane holds row (l16), cols kb..kb+1
    v2f afrag[16];
    const int  arowi = e0 + l16;
    const bool rv = arowi < E;
    const float* arow = edge_attr + (long)arowi * EDGE_DIM;
    #pragma unroll
    for (int kk = 0; kk < 16; ++kk) {
      const int kb = kk * 4 + half * 2;
      afrag[kk].x = rv ? arow[kb]     : 0.f;
      afrag[kk].y = rv ? arow[kb + 1] : 0.f;
    }

    // destination node for each of the 8 output rows this lane owns
    int drow[8];
    #pragma unroll
    for (int r = 0; r < 8; ++r) {
      const int e = e0 + r + half * 8;
      drow[r] = (full || e < E) ? (int)eidx[(long)E + e] : -1;  // dst row
    }

    for (int nt = 0; nt < 8; ++nt) {
      const int n0 = nt << 4;
      v8f acc = {};
      #pragma unroll
      for (int kk = 0; kk < 16; ++kk) {
        const int kb = kk * 4 + half * 2;
        v2f b;
        b.x = sWe[kb * NODE_DIM + n0 + l16];
        b.y = sWe[(kb + 1) * NODE_DIM + n0 + l16];
        acc = wmma_f32_4(afrag[kk], b, acc);
      }
      if (full) {  // fast path: no exec-mask churn
        #pragma unroll
        for (int r = 0; r < 8; ++r)
          atomicAdd(&sums[(long)drow[r] * NODE_DIM + n0 + l16], acc[r] + bev[nt]);
      } else {
        #pragma unroll
        for (int r = 0; r < 8; ++r)
          if (drow[r] >= 0)
            atomicAdd(&sums[(long)drow[r] * NODE_DIM + n0 + l16], acc[r] + bev[nt]);
      }
    }
  }
}

__global__ void k_cnt(const long long* __restrict__ eidx, float* cnt, int E) {
  int e = blockIdx.x * blockDim.x + threadIdx.x;
  if (e < E) atomicAdd(&cnt[(int)eidx[(long)E + e]], 1.0f);
}

// ------------------------------------------------- node GEMM h = xp @ Wg (WMMA)
__global__ void k_node_gemm(const float* __restrict__ x,
                            const float* __restrict__ sums,
                            const float* __restrict__ cnt,
                            const float* __restrict__ Wg,
                            float* __restrict__ h, int N) {
  __shared__ float sxp[4][16 * NODE_DIM];  // 4 waves * 8 KB = 32 KB
  const int lane = threadIdx.x & 31;
  const int wave = threadIdx.x >> 5;
  const int wpb  = blockDim.x >> 5;
  const int half = lane >> 4;
  const int l16  = lane & 15;
  const int nTiles = (N + 15) >> 4;
  const int tilesPerIter = gridDim.x * wpb;
  const int iters = (nTiles + tilesPerIter - 1) / tilesPerIter;

  for (int it = 0; it < iters; ++it) {
    const int t = (it * gridDim.x + blockIdx.x) * wpb + wave;
    float* xp = sxp[wave];
    if (t < nTiles) {
      const int base = t << 4;
      for (int i = lane; i < 16 * NODE_DIM; i += 32) {
        const int r = i >> 7, c = i & 127;
        const int node = base + r;
        float v = 0.f;
        if (node < N) {
          float cn = cnt[node];
          cn = cn > 1.f ? cn : 1.f;
          v = x[(long)node * NODE_DIM + c] + sums[(long)node * NODE_DIM + c] / cn;
        }
        xp[i] = v;
      }
    }
    __syncthreads();
    if (t < nTiles) {
      const int  base = t << 4;
      const bool full = (base + 16) <= N;
      for (int nt = 0; nt < 8; ++nt) {
        const int n0 = nt << 4;
        v8f acc = {};
        #pragma unroll
        for (int kk = 0; kk < 32; ++kk) {
          const int kb = kk * 4 + half * 2;
          v2f a, b;
          a.x = xp[l16 * NODE_DIM + kb];
          a.y = xp[l16 * NODE_DIM + kb + 1];
          b.x = Wg[kb * NODE_DIM + n0 + l16];
          b.y = Wg[(kb + 1) * NODE_DIM + n0 + l16];
          acc = wmma_f32_4(a, b, acc);
        }
        if (full) {
          #pragma unroll
          for (int r = 0; r < 8; ++r)
            h[(long)(base + r + half * 8) * NODE_DIM + n0 + l16] = acc[r];
        } else {
          #pragma unroll
          for (int r = 0; r < 8; ++r) {
            const int node = base + r + half * 8;
            if (node < N) h[(long)node * NODE_DIM + n0 + l16] = acc[r];
          }
        }
      }
    }
    __syncthreads();
  }
}

// ------------------------------------------------- attention scalars per node
__global__ void k_attn(const float* __restrict__ h,
                       const float* __restrict__ att_src,
                       const float* __restrict__ att_dst,
                       float* a_s, float* a_d, int N) {
  const int w = (blockIdx.x * blockDim.x + threadIdx.x) >> 5;
  const int lane = threadIdx.x & 31;
  if (w >= N) return;
  const float* hr = h + (long)w * NODE_DIM;
  float s = 0.f, d = 0.f;
  for (int c = lane; c < NODE_DIM; c += 32) {
    const float hv = hr[c];
    s += hv * att_src[c];
    d += hv * att_dst[c];
  }
  #pragma unroll
  for (int off = 16; off > 0; off >>= 1) {
    s += __shfl_down(s, off, 32);
    d += __shfl_down(d, off, 32);
  }
  if (lane == 0) { a_s[w] = s; a_d[w] = d; }
}

__global__ void k_edge_logit(const long long* __restrict__ eidx,
                             const float* __restrict__ a_s,
                             const float* __restrict__ a_d,
                             float* elog, float* segmax, int E) {
  int e = blockIdx.x * blockDim.x + threadIdx.x;
  if (e >= E) return;
  const int s = (int)eidx[e];
  const int d = (int)eidx[(long)E + e];
  const float l = leaky(a_s[s] + a_d[d]);
  elog[e] = l;
  atomicMaxF(&segmax[d], l);
}

__global__ void k_self_logit(const float* a_s, const float* a_d,
                             float* slog, float* segmax, int N) {
  int n = blockIdx.x * blockDim.x + threadIdx.x;
  if (n >= N) return;
  const float l = leaky(a_s[n] + a_d[n]);
  slog[n] = l;
  atomicMaxF(&segmax[n], l);
}

__global__ void k_self_den(const float* slog, const float* segmax,
                           float* den, int N) {
  int n = blockIdx.x * blockDim.x + threadIdx.x;
  if (n < N) den[n] = __expf(slog[n] - segmax[n]);  // first (self-loop) term
}

__global__ void k_edge_den(const long long* __restrict__ eidx,
                           const float* elog, const float* segmax,
                           float* den, int E) {
  int e = blockIdx.x * blockDim.x + threadIdx.x;
  if (e >= E) return;
  const int d = (int)eidx[(long)E + e];
  atomicAdd(&den[d], __expf(elog[e] - segmax[d]));
}

// self-loop contribution initializes the output accumulator
__global__ void k_out_self(const float* h, const float* slog,
                           const float* segmax, const float* den,
                           float* outacc, int N) {
  long i = (long)blockIdx.x * blockDim.x + threadIdx.x;
  if (i >= (long)N * NODE_DIM) return;
  const int n = (int)(i >> 7);
  const float alpha = __expf(slog[n] - segmax[n]) / den[n];
  outacc[i] = alpha * h[i];
}

// weighted scatter-add of source messages: one wave per edge, 4 channels/lane
__global__ void k_out_edge(const long long* __restrict__ eidx,
                           const float* __restrict__ elog,
                           const float* __restrict__ segmax,
                           const float* __restrict__ den,
                           const float* __restrict__ h,
                           float* outacc, int E) {
  const long tid = (long)blockIdx.x * blockDim.x + threadIdx.x;
  const int e = (int)(tid >> 5);
  if (e >= E) return;
  const int lane = (int)(tid & 31);
  const int s = (int)eidx[e];
  const int d = (int)eidx[(long)E + e];
  const float w = __expf(elog[e] - segmax[d]) / den[d];
  const float4 hv = ((const float4*)(h + (long)s * NODE_DIM))[lane];
  float* oa = outacc + (long)d * NODE_DIM + lane * 4;
  atomicAdd(oa + 0, w * hv.x);
  atomicAdd(oa + 1, w * hv.y);
  atomicAdd(oa + 2, w * hv.z);
  atomicAdd(oa + 3, w * hv.w);
}

__global__ void k_final_init(const float* bias, float* out) {
  out[threadIdx.x] = bias[threadIdx.x];  // mean(out+bias) = sum/N + bias
}

__global__ void k_final(const float* __restrict__ outacc, float* out, int N) {
  const int c = threadIdx.x;  // 128 threads
  float sum = 0.f;
  for (int n = blockIdx.x; n < N; n += gridDim.x)
    sum += outacc[(long)n * NODE_DIM + c];
  atomicAdd(&out[c], sum / (float)N);
}

// -----------------------------------------------------------------------------
extern "C" void kernel_launch(void* const* d_in, const int* in_sizes, int n_in,
                              void* d_out, int out_size, void* d_ws, size_t ws_size,
                              hipStream_t stream) {
  const float*     x         = (const float*)d_in[0];
  const long long* eidx      = (const long long*)d_in[1];  // int64 (2, E)
  const float*     edge_attr = (const float*)d_in[2];
  const float*     We        = (const float*)d_in[3];
  const float*     be        = (const float*)d_in[4];
  const float*     Wg        = (const float*)d_in[5];
  const float*     att_src   = (const float*)d_in[6];
  const float*     att_dst   = (const float*)d_in[7];
  const float*     bias      = (const float*)d_in[8];
  float*           out       = (float*)d_out;

  const int N = in_sizes[0] / NODE_DIM;
  const int E = in_sizes[2] / EDGE_DIM;

  // workspace carve-out (256B aligned chunks)
  char* ws = (char*)d_ws;
  size_t off = 0;
  auto carve = [&](size_t bytes) -> float* {
    float* p = (float*)(ws + off);
    off += (bytes + 255) & ~(size_t)255;
    return p;
  };
  float* sums   = carve((size_t)N * NODE_DIM * sizeof(float));
  float* h      = carve((size_t)N * NODE_DIM * sizeof(float));
  float* cnt    = carve((size_t)N * sizeof(float));
  float* a_s    = carve((size_t)N * sizeof(float));
  float* a_d    = carve((size_t)N * sizeof(float));
  float* segmax = carve((size_t)N * sizeof(float));
  float* den    = carve((size_t)N * sizeof(float));
  float* slog   = carve((size_t)N * sizeof(float));
  float* elog   = carve((size_t)E * sizeof(float));
  float* outacc = sums;  // sums is dead after k_node_gemm; reuse as outacc

  const long NC = (long)N * NODE_DIM;

  k_zero<<<(int)((NC + 255) / 256), 256, 0, stream>>>(sums, NC);
  k_init_node<<<(N + 255) / 256, 256, 0, stream>>>(cnt, segmax, den, N);

  k_edge_embed<<<1024, 128, 0, stream>>>(edge_attr, eidx, We, be, sums, E);
  k_cnt<<<(E + 255) / 256, 256, 0, stream>>>(eidx, cnt, E);

  k_node_gemm<<<512, 128, 0, stream>>>(x, sums, cnt, Wg, h, N);

  k_attn<<<(N * 32 + 255) / 256, 256, 0, stream>>>(h, att_src, att_dst, a_s, a_d, N);

  k_edge_logit<<<(E + 255) / 256, 256, 0, stream>>>(eidx, a_s, a_d, elog, segmax, E);
  k_self_logit<<<(N + 255) / 256, 256, 0, stream>>>(a_s, a_d, slog, segmax, N);

  k_self_den<<<(N + 255) / 256, 256, 0, stream>>>(slog, segmax, den, N);
  k_edge_den<<<(E + 255) / 256, 256, 0, stream>>>(eidx, elog, segmax, den, E);

  k_out_self<<<(int)((NC + 255) / 256), 256, 0, stream>>>(h, slog, segmax, den, outacc, N);
  k_out_edge<<<(int)(((long)E * 32 + 255) / 256), 256, 0, stream>>>(
      eidx, elog, segmax, den, h, outacc, E);

  k_final_init<<<1, 128, 0, stream>>>(bias, out);
  k_final<<<64, 128, 0, stream>>>(outacc, out, N);
}